// SingleGnnLayer_32014686224554
// MI455X (gfx1250) — compile-verified
//
#include <hip/hip_runtime.h>
#include <hip/hip_bf16.h>

typedef float v2f __attribute__((ext_vector_type(2)));
typedef float v8f __attribute__((ext_vector_type(8)));

#define N_NODES  50000
#define N_EDGES  800000
#define EMB      128
#define EDGE_DIM 16
#define HID      256          // 2*EMB
#define BN_EPS   1e-5f
#define KEEP     0.8f

// ---------------------------------------------------------------- utilities
__global__ void zero_kernel(float* __restrict__ p, long n) {
  long i = (long)blockIdx.x * blockDim.x + threadIdx.x;
  long stride = (long)gridDim.x * blockDim.x;
  for (; i < n; i += stride) p[i] = 0.0f;
}

__device__ __forceinline__ unsigned rotl32(unsigned x, int r) {
  return (x << r) | (x >> (32 - r));
}

// JAX threefry2x32, 20 rounds, key injection every 4 rounds.
__device__ __forceinline__ void threefry2x32(unsigned k0, unsigned k1,
                                             unsigned c0, unsigned c1,
                                             unsigned& o0, unsigned& o1) {
  unsigned ks[3] = {k0, k1, k0 ^ k1 ^ 0x1BD11BDAu};
  unsigned x0 = c0 + ks[0];
  unsigned x1 = c1 + ks[1];
  const int R0[4] = {13, 15, 26, 6};
  const int R1[4] = {17, 29, 16, 24};
  #pragma unroll
  for (int i = 0; i < 5; ++i) {
    #pragma unroll
    for (int j = 0; j < 4; ++j) {
      int r = (i & 1) ? R1[j] : R0[j];
      x0 += x1;
      x1 = rotl32(x1, r) ^ x0;
    }
    x0 += ks[(i + 1) % 3];
    x1 += ks[(i + 2) % 3] + (unsigned)(i + 1);
  }
  o0 = x0; o1 = x1;
}

// -------------------------------------------------- 1) edge GEMM + scatter
// msg = relu(x[src] + edge_attr @ W_edge + b_edge); aggr[dst] += msg
// One wave = 16 edges x 128 EMB (8 n-tiles x 4 k-steps of wmma f32 16x16x4).
__global__ __launch_bounds__(128)
void edge_aggr_kernel(const float* __restrict__ x,
                      const long long* __restrict__ eidx,
                      const float* __restrict__ eattr,
                      const float* __restrict__ Wedge,
                      const float* __restrict__ bedge,
                      float* __restrict__ aggr) {
  __shared__ float sW[EDGE_DIM * EMB];  // 8 KB
  __shared__ float sB[EMB];
  int tid = threadIdx.x;
  for (int i = tid; i < EDGE_DIM * EMB; i += 128) sW[i] = Wedge[i];
  if (tid < EMB) sB[tid] = bedge[tid];
  __syncthreads();

  int lane  = tid & 31;
  int wave  = tid >> 5;
  int mlo   = lane & 15;
  int khalf = lane >> 4;           // 0: rows/K 0..; 1: +2 / +8 per ISA layout
  long e0 = ((long)blockIdx.x * 4 + wave) * 16;

  // dst/src rows this lane touches in the D-fragment epilogue
  int srow[8], drow[8];
  #pragma unroll
  for (int r = 0; r < 8; ++r) {
    int m = r + 8 * khalf;
    srow[r] = (int)eidx[e0 + m];
    drow[r] = (int)eidx[(long)N_EDGES + e0 + m];
  }

  // A fragments: lane holds row mlo, (K, K+1) pairs; K base = 4s + 2*khalf
  const float* arow = eattr + (e0 + mlo) * EDGE_DIM;
  v2f afrag[4];
  #pragma unroll
  for (int s = 0; s < 4; ++s) {
    int k0 = 4 * s + 2 * khalf;
    afrag[s].x = arow[k0];
    afrag[s].y = arow[k0 + 1];
  }

  #pragma unroll
  for (int t = 0; t < 8; ++t) {
    int col = t * 16 + mlo;
    float bias = sB[col];
    v8f c;
    #pragma unroll
    for (int r = 0; r < 8; ++r) c[r] = bias;   // accumulate bias up front
    #pragma unroll
    for (int s = 0; s < 4; ++s) {
      int k0 = 4 * s + 2 * khalf;
      v2f b;
      b.x = sW[k0 * EMB + col];
      b.y = sW[(k0 + 1) * EMB + col];
      c = __builtin_amdgcn_wmma_f32_16x16x4_f32(false, afrag[s], false, b,
                                                (short)0, c, false, false);
    }
    #pragma unroll
    for (int r = 0; r < 8; ++r) {
      float v = c[r] + x[(long)srow[r] * EMB + col];
      v = fmaxf(v, 0.0f);
      atomicAdd(&aggr[(long)drow[r] * EMB + col], v);  // L2-resident
    }
  }
}

// -------------------------------------------------- 2) MLP1: h1 = z@W1 + b1
// z = (1+eps)*x + aggr built in LDS; also accumulate per-column sum/sumsq.
__global__ __launch_bounds__(256)
void mlp1_kernel(const float* __restrict__ x,
                 const float* __restrict__ aggr,
                 const float* __restrict__ eps_p,
                 const float* __restrict__ W1,
                 const float* __restrict__ b1,
                 float* __restrict__ h1,
                 float* __restrict__ stats1) {
  __shared__ float zt[16 * EMB];  // 8 KB: 16 rows x 128
  int tid = threadIdx.x;
  long row0 = (long)blockIdx.x * 16;
  float ge = 1.0f + eps_p[0];
  for (int i = tid; i < 16 * EMB; i += 256) {
    long r = i >> 7; int cc = i & 127;
    long off = (row0 + r) * EMB + cc;
    zt[i] = ge * x[off] + aggr[off];
  }
  __syncthreads();

  int lane = tid & 31, wave = tid >> 5;
  int mlo = lane & 15, khalf = lane >> 4;
  #pragma unroll
  for (int tt = 0; tt < 2; ++tt) {
    int t = wave + 8 * tt;         // 8 waves cover 16 col-tiles (256 cols)
    int ncol = t * 16 + mlo;
    float bias = b1[ncol];
    v8f c;
    #pragma unroll
    for (int r = 0; r < 8; ++r) c[r] = bias;
    #pragma unroll 8
    for (int s = 0; s < 32; ++s) { // K = 128
      int k0 = 4 * s + 2 * khalf;
      v2f a, b;
      a.x = zt[mlo * EMB + k0];
      a.y = zt[mlo * EMB + k0 + 1];
      b.x = W1[(long)k0 * HID + ncol];
      b.y = W1[(long)(k0 + 1) * HID + ncol];
      c = __builtin_amdgcn_wmma_f32_16x16x4_f32(false, a, false, b,
                                                (short)0, c, false, false);
    }
    float sum = 0.0f, sq = 0.0f;
    #pragma unroll
    for (int r = 0; r < 8; ++r) {
      int m = r + 8 * khalf;
      float v = c[r];
      h1[(row0 + m) * HID + ncol] = v;
      sum += v; sq += v * v;
    }
    sum += __shfl_xor(sum, 16, 32);    // combine the two row-halves (same col)
    sq  += __shfl_xor(sq, 16, 32);
    if (lane < 16) {
      atomicAdd(&stats1[ncol], sum);
      atomicAdd(&stats1[HID + ncol], sq);
    }
  }
}

// -------------------------------------------------- BN params: scale/shift
__global__ void bn_params_kernel(const float* __restrict__ stats,
                                 const float* __restrict__ g,
                                 const float* __restrict__ beta,
                                 float* __restrict__ bn, int C, float invN) {
  int c = blockIdx.x * blockDim.x + threadIdx.x;
  if (c >= C) return;
  float mean = stats[c] * invN;
  float var  = stats[C + c] * invN - mean * mean;
  float rstd = rsqrtf(var + BN_EPS);
  float scale = g[c] * rstd;
  bn[c]     = scale;
  bn[C + c] = beta[c] - mean * scale;
}

// -------------------------------------------------- 3) MLP2: relu(bn(h1))@W2 + b2
__global__ __launch_bounds__(256)
void mlp2_kernel(const float* __restrict__ h1,
                 const float* __restrict__ bn1,
                 const float* __restrict__ W2,
                 const float* __restrict__ b2,
                 float* __restrict__ h2,
                 float* __restrict__ stats2) {
  __shared__ float at[16 * HID];  // 16 KB: 16 rows x 256, relu(bn(h1))
  int tid = threadIdx.x;
  long row0 = (long)blockIdx.x * 16;
  for (int i = tid; i < 16 * HID; i += 256) {
    int r = i >> 8; int cc = i & 255;
    float v = h1[(row0 + r) * HID + cc];
    v = v * bn1[cc] + bn1[HID + cc];
    at[i] = fmaxf(v, 0.0f);
  }
  __syncthreads();

  int lane = tid & 31, wave = tid >> 5;
  int mlo = lane & 15, khalf = lane >> 4;
  int ncol = wave * 16 + mlo;            // 8 waves cover 128 cols
  float bias = b2[ncol];
  v8f c;
  #pragma unroll
  for (int r = 0; r < 8; ++r) c[r] = bias;
  #pragma unroll 8
  for (int s = 0; s < 64; ++s) {         // K = 256
    int k0 = 4 * s + 2 * khalf;
    v2f a, b;
    a.x = at[mlo * HID + k0];
    a.y = at[mlo * HID + k0 + 1];
    b.x = W2[(long)k0 * EMB + ncol];
    b.y = W2[(long)(k0 + 1) * EMB + ncol];
    c = __builtin_amdgcn_wmma_f32_16x16x4_f32(false, a, false, b,
                                              (short)0, c, false, false);
  }
  float sum = 0.0f, sq = 0.0f;
  #pragma unroll
  for (int r = 0; r < 8; ++r) {
    int m = r + 8 * khalf;
    float v = c[r];
    h2[(row0 + m) * EMB + ncol] = v;
    sum += v; sq += v * v;
  }
  sum += __shfl_xor(sum, 16, 32);
  sq  += __shfl_xor(sq, 16, 32);
  if (lane < 16) {
    atomicAdd(&stats2[ncol], sum);
    atomicAdd(&stats2[EMB + ncol], sq);
  }
}

// -------------------------------------------------- 4) BN2 + JAX dropout
__global__ __launch_bounds__(256)
void out_kernel(const float* __restrict__ h2,
                const float* __restrict__ bn2,
                float* __restrict__ out) {
  const long n = (long)N_NODES * EMB;
  long j = (long)blockIdx.x * 256 + threadIdx.x;
  if (j >= n) return;
  int col = (int)(j & (EMB - 1));
  float v = h2[j] * bn2[col] + bn2[EMB + col];

  // jax.random.bernoulli(key(42), 0.8, (N, EMB)) — threefry2x32 over split iota
  const long half = n / 2;
  bool lo = j < half;
  unsigned c0 = lo ? (unsigned)j : (unsigned)(j - half);
  unsigned c1 = lo ? (unsigned)(j + half) : (unsigned)j;
  unsigned o0, o1;
  threefry2x32(0u, 42u, c0, c1, o0, o1);
  unsigned bits = lo ? o0 : o1;
  float u = __uint_as_float((bits >> 9) | 0x3f800000u) - 1.0f;
  out[j] = (u < KEEP) ? v * (1.0f / KEEP) : 0.0f;
}

// ---------------------------------------------------------------- launcher
extern "C" void kernel_launch(void* const* d_in, const int* in_sizes, int n_in,
                              void* d_out, int out_size, void* d_ws, size_t ws_size,
                              hipStream_t stream) {
  const float*     x      = (const float*)d_in[0];
  const long long* eidx   = (const long long*)d_in[1];
  const float*     eattr  = (const float*)d_in[2];
  const float*     Wedge  = (const float*)d_in[3];
  const float*     bedge  = (const float*)d_in[4];
  const float*     eps    = (const float*)d_in[5];
  const float*     W1     = (const float*)d_in[6];
  const float*     b1     = (const float*)d_in[7];
  const float*     g1     = (const float*)d_in[8];
  const float*     beta1  = (const float*)d_in[9];
  const float*     W2     = (const float*)d_in[10];
  const float*     b2     = (const float*)d_in[11];
  const float*     gout   = (const float*)d_in[12];
  const float*     betao  = (const float*)d_in[13];

  float* ws     = (float*)d_ws;
  float* aggr   = ws;                                   // N*EMB
  float* h1     = aggr   + (long)N_NODES * EMB;         // N*HID
  float* h2     = h1     + (long)N_NODES * HID;         // N*EMB
  float* stats1 = h2     + (long)N_NODES * EMB;         // 2*HID
  float* stats2 = stats1 + 2 * HID;                     // 2*EMB
  float* bn1    = stats2 + 2 * EMB;                     // 2*HID
  float* bn2    = bn1    + 2 * HID;                     // 2*EMB
  float* out    = (float*)d_out;

  zero_kernel<<<512, 256, 0, stream>>>(aggr, (long)N_NODES * EMB);
  zero_kernel<<<2, 256, 0, stream>>>(stats1, 2 * HID + 2 * EMB);

  edge_aggr_kernel<<<N_EDGES / 64, 128, 0, stream>>>(x, eidx, eattr, Wedge, bedge, aggr);

  mlp1_kernel<<<N_NODES / 16, 256, 0, stream>>>(x, aggr, eps, W1, b1, h1, stats1);
  bn_params_kernel<<<1, 256, 0, stream>>>(stats1, g1, beta1, bn1, HID, 1.0f / N_NODES);

  mlp2_kernel<<<N_NODES / 16, 256, 0, stream>>>(h1, bn1, W2, b2, h2, stats2);
  bn_params_kernel<<<1, 128, 0, stream>>>(stats2, gout, betao, bn2, EMB, 1.0f / N_NODES);

  long n = (long)N_NODES * EMB;
  out_kernel<<<(int)((n + 255) / 256), 256, 0, stream>>>(h2, bn2, out);
}